// NeuroBLASTModel_47270410060067
// MI455X (gfx1250) — compile-verified
//
#include <hip/hip_runtime.h>
#include <math.h>

// CDNA5 / gfx1250. fp32 throughout (reference is fp32; workload is HBM-bound:
// ~268 MB cache copy + ~300 MB strided weight slice >> 3.76 GFLOP of f32 WMMA).

typedef __attribute__((ext_vector_type(2))) float v2f;
typedef __attribute__((ext_vector_type(4))) float v4f;
typedef __attribute__((ext_vector_type(8))) float v8f;

#define B_ 128
#define C_ 1024
#define O_ 2048
#define KPAD 8192          // 1024 c * 8 (7 taps + 1 zero pad) -> K multiple of 4
#define CACHE_ROWS 32768
#define TRASH (CACHE_ROWS - 1)
#define EPS_ 1e-6f
#define NSPLIT 4           // K-dim split for occupancy (weights still read once)
#define CHUNKS_PER_SPLIT (64 / NSPLIT)

// ---------------- K1: new_cache = conv_cache (bulk copy, non-temporal) ----------------
__global__ void copy_cache_kernel(const v4f* __restrict__ src,
                                  v4f* __restrict__ dst, int n4) {
    int i = blockIdx.x * blockDim.x + threadIdx.x;
    if (i < n4) {
        v4f v = __builtin_nontemporal_load(src + i);
        __builtin_nontemporal_store(v, dst + i);
    }
}

// ------- K2: rmsnorm(x), gather past rows, pack Xp, scatter to cache -------
__global__ void __launch_bounds__(256)
prep_kernel(const float* __restrict__ x,
            const int* __restrict__ positions,
            const int* __restrict__ block_tables,
            const int* __restrict__ slot_mapping,
            const float* __restrict__ conv_cache,
            const float* __restrict__ w_in,
            float* __restrict__ Xp,
            float* __restrict__ cache_out) {
    const int b = blockIdx.x;
    const int t = threadIdx.x;
    __shared__ float red[256];
    __shared__ int s_slots[6];
    __shared__ int s_cur, s_ok;
    __shared__ float s_rinv;

    if (t < 6) {
        int pp = positions[b] - 6 + t;
        if (pp < 0) pp = 0;
        int phys = block_tables[b * (CACHE_ROWS / 16) + (pp >> 4)];
        long slot = (long)phys * 16 + (pp & 15);
        if (slot < 0 || slot >= CACHE_ROWS) slot = TRASH;
        s_slots[t] = (int)slot;
    }
    if (t == 6) {
        int cur = slot_mapping[b];
        if (cur < 0 || cur >= CACHE_ROWS) cur = TRASH;
        int ok = 1;   // deterministic "last batch wins" on duplicate slots
        for (int bb = b + 1; bb < B_; ++bb) {
            int c2 = slot_mapping[bb];
            if (c2 < 0 || c2 >= CACHE_ROWS) c2 = TRASH;
            if (c2 == cur) { ok = 0; break; }
        }
        s_cur = cur; s_ok = ok;
    }

    float xv[4];
    float ss = 0.f;
    #pragma unroll
    for (int i = 0; i < 4; ++i) {
        int c = t + i * 256;
        float v = x[(size_t)b * C_ + c];
        xv[i] = v;
        ss += v * v;
    }
    red[t] = ss;
    __syncthreads();
    for (int s = 128; s > 0; s >>= 1) {
        if (t < s) red[t] += red[t + s];
        __syncthreads();
    }
    if (t == 0) s_rinv = rsqrtf(red[0] / (float)C_ + EPS_);
    __syncthreads();

    const float rinv = s_rinv;
    const int cur = s_cur, ok = s_ok;
    #pragma unroll
    for (int i = 0; i < 4; ++i) {
        int c = t + i * 256;
        float y = w_in[c] * xv[i] * rinv;
        float* xp = Xp + (size_t)b * KPAD + c * 8;
        #pragma unroll
        for (int kw = 0; kw < 6; ++kw)
            xp[kw] = conv_cache[(size_t)s_slots[kw] * C_ + c];
        xp[6] = y;       // current token
        xp[7] = 0.f;     // K padding
        if (ok) cache_out[(size_t)cur * C_ + c] = y;
    }
}

// ---------------- K3: Ypart[ks] += Xp x W^T via WMMA ----------------
// Block = (n_tile, k_split): 16-wide N-tile, all 128 M rows (8 waves x 16x16),
// 16 channel-chunks of the K range. Weight kh=3 slice streamed non-temporally
// (each element fetched exactly once device-wide), double-buffered in LDS so
// the strided HBM weight stream overlaps the WMMA burst.
__global__ void __launch_bounds__(256)
gemm_kernel(const float* __restrict__ Xp,
            const float* __restrict__ conv_weight,
            float* __restrict__ Ypart) {
    const int n_tile = blockIdx.x & (O_ / 16 - 1);
    const int ks     = blockIdx.x >> 7;          // 0..NSPLIT-1
    const int n0     = n_tile * 16;
    const int cc0    = ks * CHUNKS_PER_SPLIT;

    const int tid  = threadIdx.x;
    const int wave = tid >> 5;
    const int lane = tid & 31;
    const int half = lane >> 4;     // K half-select per ISA A/B layout
    const int l16  = lane & 15;

    __shared__ float Bt[2][16 * 130];  // [n][k_local], pitch 130: no bank conflicts

    v8f acc0 = {0.f,0.f,0.f,0.f,0.f,0.f,0.f,0.f};
    v8f acc1 = {0.f,0.f,0.f,0.f,0.f,0.f,0.f,0.f};

    const int o_l = tid >> 4;       // 0..15 (cooperative weight-tile load)
    const int c_l = tid & 15;

    // kh=3 slice: 7 contiguous floats at offset 3*7=21 within each 49-elem kernel
    float wreg[7];
    {
        int c = cc0 * 16 + c_l;
        const float* wp = conv_weight + ((size_t)(n0 + o_l) * C_ + c) * 49 + 21;
        #pragma unroll
        for (int kw = 0; kw < 7; ++kw) wreg[kw] = __builtin_nontemporal_load(wp + kw);
    }
    {
        float* bt = &Bt[0][o_l * 130 + c_l * 8];
        #pragma unroll
        for (int kw = 0; kw < 7; ++kw) bt[kw] = wreg[kw];
        bt[7] = 0.f;
    }
    __syncthreads();

    for (int i = 0; i < CHUNKS_PER_SPLIT; ++i) {
        const int cc = cc0 + i;
        if (i + 1 < CHUNKS_PER_SPLIT) {           // prefetch next tile into regs
            int c = (cc + 1) * 16 + c_l;
            const float* wp = conv_weight + ((size_t)(n0 + o_l) * C_ + c) * 49 + 21;
            #pragma unroll
            for (int kw = 0; kw < 7; ++kw) wreg[kw] = __builtin_nontemporal_load(wp + kw);
        }

        const float* arow = Xp + (size_t)(wave * 16 + l16) * KPAD + cc * 128 + half * 2;
        const float* brow = &Bt[i & 1][l16 * 130 + half * 2];
        #pragma unroll
        for (int kk = 0; kk < 32; kk += 2) {      // dual accumulators for pipelining
            v2f a0 = *(const v2f*)(arow + (kk    ) * 4);
            v2f b0 = *(const v2f*)(brow + (kk    ) * 4);
            v2f a1 = *(const v2f*)(arow + (kk + 1) * 4);
            v2f b1 = *(const v2f*)(brow + (kk + 1) * 4);
            acc0 = __builtin_amdgcn_wmma_f32_16x16x4_f32(false, a0, false, b0,
                                                         (short)0, acc0, false, false);
            acc1 = __builtin_amdgcn_wmma_f32_16x16x4_f32(false, a1, false, b1,
                                                         (short)0, acc1, false, false);
        }

        if (i + 1 < CHUNKS_PER_SPLIT) {           // fill the other buffer
            float* bt = &Bt[(i + 1) & 1][o_l * 130 + c_l * 8];
            #pragma unroll
            for (int kw = 0; kw < 7; ++kw) bt[kw] = wreg[kw];
            bt[7] = 0.f;
        }
        __syncthreads();
    }

    v8f acc = acc0 + acc1;
    // C/D layout: VGPR r -> M = r (lanes 0-15) / r+8 (lanes 16-31), N = lane&15
    float* yp = Ypart + (size_t)ks * B_ * O_;
    #pragma unroll
    for (int r = 0; r < 8; ++r) {
        int row = wave * 16 + r + half * 8;
        int col = n0 + l16;
        yp[(size_t)row * O_ + col] = acc[r];
    }
}

// ------- K4: reduce K-splits, gate softmax * val, rmsnorm, residual -> out -------
__global__ void __launch_bounds__(256)
epilogue_kernel(const float* __restrict__ x,
                const float* __restrict__ Ypart,
                const float* __restrict__ w_out,
                float* __restrict__ out) {
    const int b = blockIdx.x;
    const int t = threadIdx.x;
    __shared__ float red[256];
    __shared__ float s_bcast;

    float g[4], v[4];
    #pragma unroll
    for (int i = 0; i < 4; ++i) {
        int c = t + i * 256;
        float sg = 0.f, sv = 0.f;
        #pragma unroll
        for (int ks = 0; ks < NSPLIT; ++ks) {
            const float* yp = Ypart + (size_t)ks * B_ * O_ + (size_t)b * O_;
            sg += yp[c];
            sv += yp[C_ + c];
        }
        g[i] = sg;
        v[i] = sv;
    }
    // softmax max
    red[t] = fmaxf(fmaxf(g[0], g[1]), fmaxf(g[2], g[3]));
    __syncthreads();
    for (int s = 128; s > 0; s >>= 1) {
        if (t < s) red[t] = fmaxf(red[t], red[t + s]);
        __syncthreads();
    }
    if (t == 0) s_bcast = red[0];
    __syncthreads();
    const float mx = s_bcast;
    __syncthreads();
    // softmax sum
    float e[4], es = 0.f;
    #pragma unroll
    for (int i = 0; i < 4; ++i) { e[i] = __expf(g[i] - mx); es += e[i]; }
    red[t] = es;
    __syncthreads();
    for (int s = 128; s > 0; s >>= 1) {
        if (t < s) red[t] += red[t + s];
        __syncthreads();
    }
    if (t == 0) s_bcast = red[0];
    __syncthreads();
    const float inv_sum = 1.f / s_bcast;
    __syncthreads();
    // y = val * softmax(gate); rms reduce
    float y[4], ss = 0.f;
    #pragma unroll
    for (int i = 0; i < 4; ++i) { y[i] = v[i] * e[i] * inv_sum; ss += y[i] * y[i]; }
    red[t] = ss;
    __syncthreads();
    for (int s = 128; s > 0; s >>= 1) {
        if (t < s) red[t] += red[t + s];
        __syncthreads();
    }
    if (t == 0) s_bcast = rsqrtf(red[0] / (float)C_ + EPS_);
    __syncthreads();
    const float rinv = s_bcast;
    #pragma unroll
    for (int i = 0; i < 4; ++i) {
        int c = t + i * 256;
        out[(size_t)b * C_ + c] = x[(size_t)b * C_ + c] + w_out[c] * y[i] * rinv;
    }
}

extern "C" void kernel_launch(void* const* d_in, const int* in_sizes, int n_in,
                              void* d_out, int out_size, void* d_ws, size_t ws_size,
                              hipStream_t stream) {
    const float* x           = (const float*)d_in[0];
    const int*   positions   = (const int*)d_in[1];
    const int*   block_tables= (const int*)d_in[2];
    const int*   slot_mapping= (const int*)d_in[3];
    const float* conv_cache  = (const float*)d_in[4];
    const float* conv_weight = (const float*)d_in[5];
    const float* w_in        = (const float*)d_in[6];
    const float* w_out       = (const float*)d_in[7];

    float* out       = (float*)d_out;                  // [B, C]
    float* cache_out = out + (size_t)B_ * C_;          // [CACHE_ROWS, C]

    float* Xp    = (float*)d_ws;                       // [B, KPAD]          (4 MB)
    float* Ypart = Xp + (size_t)B_ * KPAD;             // [NSPLIT, B, O]     (4 MB)

    const int n4 = (CACHE_ROWS * C_) / 4;
    copy_cache_kernel<<<n4 / 256, 256, 0, stream>>>((const v4f*)conv_cache,
                                                    (v4f*)cache_out, n4);
    prep_kernel<<<B_, 256, 0, stream>>>(x, positions, block_tables, slot_mapping,
                                        conv_cache, w_in, Xp, cache_out);
    gemm_kernel<<<(O_ / 16) * NSPLIT, 256, 0, stream>>>(Xp, conv_weight, Ypart);
    epilogue_kernel<<<B_, 256, 0, stream>>>(x, Ypart, w_out, out);
}